// GCN_24842090840539
// MI455X (gfx1250) — compile-verified
//
#include <hip/hip_runtime.h>

#define NN 50000
#define F  128
#define MT 3125   // 50000 / 16 row tiles (exact)

typedef __attribute__((ext_vector_type(2))) float v2f;
typedef __attribute__((ext_vector_type(8))) float v8f;

// ---------------- degree / normalization ----------------

__global__ void GCN_deg_init(float* deg) {
    int i = blockIdx.x * blockDim.x + threadIdx.x;
    if (i < NN) deg[i] = 1.0f;                    // self-loop contribution
}

__global__ void GCN_deg_accum(const long long* __restrict__ dst, float* deg, int E) {
    int e = blockIdx.x * blockDim.x + threadIdx.x;
    if (e < E) atomicAdd(&deg[(int)dst[e]], 1.0f);
}

__global__ void GCN_dinv(float* deg) {
    int i = blockIdx.x * blockDim.x + threadIdx.x;
    if (i < NN) { float d = deg[i]; deg[i] = (d > 0.f) ? rsqrtf(d) : 0.f; }
}

__global__ void GCN_zero(float* p, int n) {
    int i = blockIdx.x * blockDim.x + threadIdx.x;
    if (i < n) p[i] = 0.f;
}

// ---------------- GEMM: H = X @ W  (f32 WMMA 16x16x4) ----------------
// One wave computes a 16-row x 128-col output tile. W staged in LDS.

__global__ __launch_bounds__(256) void GCN_gemm(const float* __restrict__ X,
                                                const float* __restrict__ W,
                                                float* __restrict__ H) {
    __shared__ float sW[F * F];                   // 64 KB (WGP has 320 KB LDS)
    for (int i = threadIdx.x * 4; i < F * F; i += 256 * 4)
        *(float4*)(sW + i) = *(const float4*)(W + i);
    __syncthreads();

    const int wave  = threadIdx.x >> 5;
    const int lane  = threadIdx.x & 31;
    const int mtile = blockIdx.x * 8 + wave;
    if (mtile >= MT) return;

    const int lo = lane & 15;                     // row (A) / col (B) within tile
    const int hi = lane >> 4;                     // half-select
    const int kh = hi << 1;                       // K offset 0 or 2

    const float* Xr = X + (mtile * 16 + lo) * F;

    v8f acc[8] = {};
    for (int k = 0; k < F; k += 4) {
        v2f a;
        a.x = Xr[k + kh];
        a.y = Xr[k + kh + 1];
        const float* Wk = sW + (k + kh) * F + lo;
#pragma unroll
        for (int n = 0; n < 8; ++n) {
            v2f b;
            b.x = Wk[n * 16];
            b.y = Wk[F + n * 16];
            acc[n] = __builtin_amdgcn_wmma_f32_16x16x4_f32(
                false, a, false, b, (short)0, acc[n], false, false);
        }
    }

    // C/D layout: VGPR r -> row r + 8*hi, col = lo (+ 16*n)
    float* Hr = H + (mtile * 16) * F;
#pragma unroll
    for (int r = 0; r < 8; ++r) {
        const int m = r + hi * 8;
#pragma unroll
        for (int n = 0; n < 8; ++n)
            Hr[m * F + n * 16 + lo] = acc[n][r];
    }
}

// ---------------- edge aggregation: one wave per edge ----------------

__global__ __launch_bounds__(256) void GCN_agg(const long long* __restrict__ src,
                                               const long long* __restrict__ dst,
                                               const float* __restrict__ dinv,
                                               const float* __restrict__ H,
                                               float* A, int E) {
    int gid  = blockIdx.x * blockDim.x + threadIdx.x;
    int e    = gid >> 5;
    int lane = gid & 31;
    if (e >= E) return;
    int s = (int)src[e];
    int d = (int)dst[e];
    float w = dinv[s] * dinv[d];
    const float4 v = *(const float4*)(H + s * F + lane * 4);
    float* out = A + d * F + lane * 4;
    atomicAdd(out + 0, v.x * w);
    atomicAdd(out + 1, v.y * w);
    atomicAdd(out + 2, v.z * w);
    atomicAdd(out + 3, v.w * w);
}

// ------------- finalize: out = relu(agg + dinv^2 * h + b) -------------

__global__ __launch_bounds__(256) void GCN_finalize(const float* A, const float* Hm,
                                                    const float* __restrict__ dinv,
                                                    const float* __restrict__ b,
                                                    float* O) {
    int gid  = blockIdx.x * blockDim.x + threadIdx.x;
    int node = gid >> 5;
    int j    = (gid & 31) * 4;
    if (node >= NN) return;
    float di = dinv[node];
    float w  = di * di;
    const float4 a  = *(const float4*)(A + node * F + j);
    const float4 h  = *(const float4*)(Hm + node * F + j);
    const float4 bb = *(const float4*)(b + j);
    float4 o;
    o.x = fmaxf(a.x + w * h.x + bb.x, 0.f);
    o.y = fmaxf(a.y + w * h.y + bb.y, 0.f);
    o.z = fmaxf(a.z + w * h.z + bb.z, 0.f);
    o.w = fmaxf(a.w + w * h.w + bb.w, 0.f);
    *(float4*)(O + node * F + j) = o;
}

// ---------------- driver ----------------

extern "C" void kernel_launch(void* const* d_in, const int* in_sizes, int n_in,
                              void* d_out, int out_size, void* d_ws, size_t ws_size,
                              hipStream_t stream) {
    const float*     x   = (const float*)d_in[0];
    const long long* ei  = (const long long*)d_in[1];   // int64 per reference
    const float*     W1  = (const float*)d_in[2];
    const float*     b1  = (const float*)d_in[3];
    const float*     W2  = (const float*)d_in[4];
    const float*     b2  = (const float*)d_in[5];
    float*           out = (float*)d_out;

    const int E = in_sizes[1] / 2;
    const long long* src = ei;
    const long long* dst = ei + E;

    char*  ws   = (char*)d_ws;
    float* dinv = (float*)ws;                                     // NN floats
    float* bufA = (float*)(ws + ((NN * 4 + 255) / 256) * 256);    // NN*F floats
    float* bufB = bufA + NN * F;                                  // NN*F floats

    const int T = 256;
    // normalization
    GCN_deg_init <<<(NN + T - 1) / T, T, 0, stream>>>(dinv);
    GCN_deg_accum<<<(E + T - 1) / T,  T, 0, stream>>>(dst, dinv, E);
    GCN_dinv     <<<(NN + T - 1) / T, T, 0, stream>>>(dinv);

    const int gemmBlocks = (MT + 7) / 8;
    const int aggBlocks  = (E * 32 + T - 1) / T;
    const int finBlocks  = (NN * 32 + T - 1) / T;

    // layer 1: h1 = x@W1 (bufA); agg -> bufB; a1 = relu(...) in-place in bufB
    GCN_gemm    <<<gemmBlocks, T, 0, stream>>>(x, W1, bufA);
    GCN_zero    <<<(NN * F + T - 1) / T, T, 0, stream>>>(bufB, NN * F);
    GCN_agg     <<<aggBlocks, T, 0, stream>>>(src, dst, dinv, bufA, bufB, E);
    GCN_finalize<<<finBlocks, T, 0, stream>>>(bufB, bufA, dinv, b1, bufB);

    // layer 2: h2 = a1@W2 (bufA); agg -> bufB; out = relu(...)
    GCN_gemm    <<<gemmBlocks, T, 0, stream>>>(bufB, W2, bufA);
    GCN_zero    <<<(NN * F + T - 1) / T, T, 0, stream>>>(bufB, NN * F);
    GCN_agg     <<<aggBlocks, T, 0, stream>>>(src, dst, dinv, bufA, bufB, E);
    GCN_finalize<<<finBlocks, T, 0, stream>>>(bufB, bufA, dinv, b2, out);
}